// DifferentiableCIndexLoss_89593017795085
// MI455X (gfx1250) — compile-verified
//
#include <hip/hip_runtime.h>

// CDNA5 / gfx1250, wave32.
typedef __attribute__((ext_vector_type(2))) float v2f;
typedef __attribute__((ext_vector_type(8))) float v8f;

// sigmoid(d/SIGMA) with SIGMA = 0.1:
//   sigmoid(d/SIGMA) = 0.5 + 0.5 * tanh(d / (2*SIGMA)) = 0.5 + 0.5*tanh(5*d)
// The 5x scale is linear, so it is folded into the WMMA operands:
//   A = [1, -5*risk_i], B = [5*risk_j; 1]  =>  D = 5*(risk_j - risk_i)
#define HALF_INV_SIGMA        (5.0f)
#define NEG_LOG2E_OVER_SIGMA (-14.4269504088896340736f)

__device__ __forceinline__ float sigmoid_from_scaled(float d5)
{
#if __has_builtin(__builtin_amdgcn_tanhf)
    // One quarter-rate transcendental per element (v_tanh_f32).
    const float t = __builtin_amdgcn_tanhf(d5);
    return __builtin_fmaf(0.5f, t, 0.5f);
#else
    // Fallback: sigmoid(x) with x = d/SIGMA = 2*d5 -> exp2(-2*d5*log2e)
    const float e = __builtin_amdgcn_exp2f(d5 * (2.0f * -1.4426950408889634f));
    return __builtin_amdgcn_rcpf(1.0f + e);
#endif
}

// One wave (32 threads) per block. Each block handles one 16-row tile of i
// (blockIdx.x) and a chunk of column tiles (blockIdx.y). For each 16x16 tile,
// a single v_wmma_f32_16x16x4_f32 computes the rank-2 outer-product
// 5*(risk[j] - risk[i]) for all 256 pairs on the matrix pipe; the VALU then
// applies the hardware tanh-sigmoid + mask, co-executing with the matrix op.
__global__ __launch_bounds__(32) void cidx_tiles_kernel(
    const float* __restrict__ risk,
    const float* __restrict__ times,
    const int*   __restrict__ events,
    float* __restrict__ partials,     // [2 * gridDim.x * gridDim.y]
    int colTilesPerWave)
{
    const int lane  = threadIdx.x;    // 0..31
    const int laneN = lane & 15;      // column within tile for B/C/D layout
    const int hi    = lane >> 4;      // 0: lanes 0-15, 1: lanes 16-31
    const int i0    = blockIdx.x * 16;               // row-tile base
    const int jt0   = blockIdx.y * colTilesPerWave;  // first column tile

    // Loop-invariant A matrix (16x4 f32, 2 VGPRs):
    //   lanes 0-15 hold K=0 (a.x) and K=1 (a.y) for row M=lane.
    //   Column K0 = ones, K1 = -5*risk_i, K2 = K3 = 0 (lanes 16-31).
    v2f a;
    a.x = hi ? 0.0f : 1.0f;
    a.y = hi ? 0.0f : (-HALF_INV_SIGMA) * risk[i0 + laneN];

    // Loop-invariant per-lane row data: this lane's 8 M values are v + 8*hi.
    float ti[8];
    float ev[8];
#pragma unroll
    for (int v = 0; v < 8; ++v) {
        const int m = i0 + v + 8 * hi;
        ti[v] = times[m];
        ev[v] = (events[m] == 1) ? 1.0f : 0.0f;
    }

    float loss = 0.0f;
    float cnt  = 0.0f;

    for (int t = 0; t < colTilesPerWave; ++t) {
        const int j0 = (jt0 + t) * 16;
        const float rj = risk[j0 + laneN];   // B row 0 source: risk[j]
        const float tj = times[j0 + laneN];  // needed by all lanes for mask

        // B matrix (4x16 f32, 2 VGPRs): row K=0 = 5*risk_j, row K=1 = ones.
        // Rows K=2,3 (lanes 16-31) multiply A's zero columns -> irrelevant.
        v2f b;
        b.x = hi ? 0.0f : HALF_INV_SIGMA * rj;
        b.y = hi ? 0.0f : 1.0f;

        v8f c = {};
        // D(M,N) = 5*(risk[j0+N] - risk[i0+M]): full tile, scale pre-folded.
        v8f d = __builtin_amdgcn_wmma_f32_16x16x4_f32(
            /*neg_a=*/false, a, /*neg_b=*/false, b,
            /*c_mod=*/(short)0, c, /*reuse_a=*/false, /*reuse_b=*/false);

#pragma unroll
        for (int v = 0; v < 8; ++v) {
            const float m = (ti[v] < tj) ? ev[v] : 0.0f;    // mask as 0/1 float
            const float s = sigmoid_from_scaled(d[v]);
            loss = __builtin_fmaf(m, s, loss);
            cnt += m;
        }
    }

    // Wave32 tree reduction (deterministic).
#pragma unroll
    for (int off = 16; off > 0; off >>= 1) {
        loss += __shfl_xor(loss, off, 32);
        cnt  += __shfl_xor(cnt,  off, 32);
    }

    if (lane == 0) {
        const int bid = blockIdx.y * gridDim.x + blockIdx.x;
        partials[2 * bid]     = loss;
        partials[2 * bid + 1] = cnt;
    }
}

// Single-wave final reduction in double (fixed order -> deterministic).
__global__ __launch_bounds__(32) void cidx_reduce_kernel(
    const float* __restrict__ partials, int nPart, float* __restrict__ out)
{
    double l = 0.0, c = 0.0;
    for (int k = (int)threadIdx.x; k < nPart; k += 32) {
        l += (double)partials[2 * k];
        c += (double)partials[2 * k + 1];
    }
#pragma unroll
    for (int off = 16; off > 0; off >>= 1) {
        l += __shfl_xor(l, off, 32);
        c += __shfl_xor(c, off, 32);
    }
    if (threadIdx.x == 0) {
        out[0] = (float)(l / (c + 1e-6));
    }
}

extern "C" void kernel_launch(void* const* d_in, const int* in_sizes, int n_in,
                              void* d_out, int out_size, void* d_ws, size_t ws_size,
                              hipStream_t stream)
{
    (void)n_in; (void)out_size; (void)ws_size;

    const float* risk   = (const float*)d_in[0];
    const float* times  = (const float*)d_in[1];
    const int*   events = (const int*)d_in[2];
    float* out      = (float*)d_out;
    float* partials = (float*)d_ws;

    const int B = in_sizes[0];                 // 16384
    const int rowTiles  = B / 16;              // 1024
    const int colChunks = 8;
    const int colTilesPerWave = rowTiles / colChunks; // 128 column tiles / wave

    dim3 grid(rowTiles, colChunks);
    cidx_tiles_kernel<<<grid, 32, 0, stream>>>(risk, times, events,
                                               partials, colTilesPerWave);

    const int nPart = rowTiles * colChunks;    // 8192 partial pairs
    cidx_reduce_kernel<<<1, 32, 0, stream>>>(partials, nPart, out);
}